// HotPoint_82583631168083
// MI455X (gfx1250) — compile-verified
//
#include <hip/hip_runtime.h>
#include <hip/hip_bf16.h>

typedef __attribute__((ext_vector_type(2))) float v2f;
typedef __attribute__((ext_vector_type(4))) float v4f;
typedef __attribute__((ext_vector_type(8))) float v8f;

// ---------------------------------------------------------------------------
// Kernel 1: fused  s1/s2 = (relu(text @ W0^T + b0)) . {w_a, w_b}
//   text : (8192, 768) fp32   (M x K)
//   W0   : (128, 768)  fp32   (N x K)
//   Each block: one 16-row M tile. 8 waves, wave w -> N tile [16w, 16w+16).
//   192 chained V_WMMA_F32_16X16X4_F32 per wave over K=768.
// ---------------------------------------------------------------------------
__global__ __launch_bounds__(256)
void fused_gemm_reduce(const float* __restrict__ text,
                       const float* __restrict__ W0,
                       const float* __restrict__ b0,
                       const float* __restrict__ W1,   // 256 floats: [w_a | w_b]
                       float* __restrict__ s1,         // (8192)
                       float* __restrict__ s2)         // (8192)
{
    __shared__ float lds_s1[8][16];
    __shared__ float lds_s2[8][16];

    const int tid  = threadIdx.x;
    const int wave = tid >> 5;        // 0..7  -> N tile
    const int lane = tid & 31;
    const int l15  = lane & 15;
    const int hi   = lane >> 4;       // 0: K={0,1}, 1: K={2,3}

    const int Mbase = blockIdx.x * 16;          // 512 blocks -> 8192 rows
    const int mrow  = Mbase + l15;              // A row (per ISA f32 A layout)
    const int nrow  = wave * 16 + l15;          // B column == W0 row

    const float* aptr = text + (size_t)mrow * 768 + 2 * hi;
    const float* bptr = W0   + (size_t)nrow * 768 + 2 * hi;

    v8f c = {0.f, 0.f, 0.f, 0.f, 0.f, 0.f, 0.f, 0.f};

    #pragma unroll 4
    for (int k = 0; k < 768; k += 4) {
        v2f a = *(const v2f*)(aptr + k);   // lane-local K pair, 8B aligned
        v2f b = *(const v2f*)(bptr + k);
        // D = A(16x4,f32) * B(4x16,f32) + C(16x16,f32)
        c = __builtin_amdgcn_wmma_f32_16x16x4_f32(false, a, false, b,
                                                  (short)0, c, false, false);
    }

    const float bias = b0[nrow];
    const float wa   = W1[nrow];
    const float wb   = W1[128 + nrow];

    // C/D layout: VGPR r holds (M = r + 8*hi, N = l15).
    #pragma unroll
    for (int r = 0; r < 8; ++r) {
        float v  = fmaxf(c[r] + bias, 0.0f);   // relu(h) element
        float p1 = v * wa;
        float p2 = v * wb;
        // reduce the 16 N-lanes inside each half-wave (masks < 16 stay in half)
        #pragma unroll
        for (int m = 8; m >= 1; m >>= 1) {
            p1 += __shfl_xor(p1, m, 32);
            p2 += __shfl_xor(p2, m, 32);
        }
        if (l15 == 0) {
            const int mloc = r + hi * 8;
            lds_s1[wave][mloc] = p1;
            lds_s2[wave][mloc] = p2;
        }
    }
    __syncthreads();

    // combine the 8 N-tiles (waves) and store the per-row scalars
    if (tid < 16) {
        float a1 = 0.f, a2 = 0.f;
        #pragma unroll
        for (int w = 0; w < 8; ++w) { a1 += lds_s1[w][tid]; a2 += lds_s2[w][tid]; }
        s1[Mbase + tid] = a1;
        s2[Mbase + tid] = a2;
    }
}

// ---------------------------------------------------------------------------
// Kernel 2: out[b,i,j] = sigmoid(s1[b,i] + s2[b,j] + b1)   (32 MB stream-out)
// One thread -> 4 consecutive j (b128 store).
// ---------------------------------------------------------------------------
__global__ __launch_bounds__(256)
void outer_sigmoid(const float* __restrict__ s1,
                   const float* __restrict__ s2,
                   const float* __restrict__ b1,
                   float* __restrict__ out)
{
    const int gid = blockIdx.x * 256 + threadIdx.x;
    const int row = gid >> 8;            // b*1024 + i   (0..8191)
    const int j0  = (gid & 255) << 2;    // 0..1020 step 4
    const int b   = row >> 10;

    const float x1 = s1[row] + b1[0];
    const v4f  y2  = *(const v4f*)(s2 + b * 1024 + j0);

    v4f o;
    #pragma unroll
    for (int t = 0; t < 4; ++t) {
        const float x = x1 + y2[t];
        o[t] = 1.0f / (1.0f + __expf(-x));
    }
    *(v4f*)(out + (size_t)row * 1024 + j0) = o;
}

// ---------------------------------------------------------------------------
// Inputs (setup_inputs order): text, mask(unused), W0, b0, W1, b1
// ---------------------------------------------------------------------------
extern "C" void kernel_launch(void* const* d_in, const int* in_sizes, int n_in,
                              void* d_out, int out_size, void* d_ws, size_t ws_size,
                              hipStream_t stream) {
    const float* text = (const float*)d_in[0];
    const float* W0   = (const float*)d_in[2];
    const float* b0   = (const float*)d_in[3];
    const float* W1   = (const float*)d_in[4];
    const float* b1   = (const float*)d_in[5];
    float* out = (float*)d_out;

    float* s1 = (float*)d_ws;          // 8192 floats
    float* s2 = s1 + 8192;             // 8192 floats

    fused_gemm_reduce<<<dim3(512), dim3(256), 0, stream>>>(text, W0, b0, W1, s1, s2);

    // 8 * 1024 * 1024 outputs / (256 threads * 4 per thread) = 8192 blocks
    outer_sigmoid<<<dim3(8192), dim3(256), 0, stream>>>(s1, s2, b1, out);
}